// FpsTrainModel_89386859364998
// MI455X (gfx1250) — compile-verified
//
#include <hip/hip_runtime.h>

#define B_          8
#define N_          65536
#define SPER_       4096
#define S_          32768
#define OUT_        256
#define THREADS_FPS 1024
#define PTS_PER     (N_ / THREADS_FPS)   // 64 points per thread, register-resident mind[]

typedef __attribute__((ext_vector_type(2))) float v2f;
typedef __attribute__((ext_vector_type(8))) float v8f;

__device__ __forceinline__ void argmax_combine(float& v, int& i, float ov, int oi) {
    if (ov > v || (ov == v && oi < i)) { v = ov; i = oi; }
}

// ---------------------------------------------------------------------------
// Kernel 1: farthest-point sampling. One workgroup per cloud (8 WGs total).
// mind[] kept in VGPRs (64 per thread); x streamed from L2 each step.
// Writes r_out[s] = (float)(global index) for all S samples.
// ---------------------------------------------------------------------------
__global__ __launch_bounds__(THREADS_FPS) void fps_kernel(const float* __restrict__ x,
                                                          float* __restrict__ r_out) {
    const int b    = blockIdx.x;
    const int tid  = threadIdx.x;
    const int lane = tid & 31;
    const int wave = tid >> 5;
    const size_t base = (size_t)b * N_;

    __shared__ float swv[32];
    __shared__ int   swi[32];
    __shared__ int   s_win;

    float mind[PTS_PER];

    // start point = point 0 of this cloud
    float cx = x[base * 3 + 0];
    float cy = x[base * 3 + 1];
    float cz = x[base * 3 + 2];

    if (tid == 0) r_out[(size_t)b * SPER_ + 0] = (float)base;

    float best = -1.0f; int besti = 0;
    #pragma unroll
    for (int i = 0; i < PTS_PER; ++i) {
        int p = tid + i * THREADS_FPS;
        const float* xp = x + (base + (size_t)p) * 3;
        float dx = xp[0] - cx, dy = xp[1] - cy, dz = xp[2] - cz;
        float d = dx * dx + dy * dy + dz * dz;
        mind[i] = d;
        if (d > best) { best = d; besti = p; }
    }

    for (int s = 1; s < SPER_; ++s) {
        // ---- argmax over the workgroup (val, lowest-index tie-break) ----
        float v = best; int ix = besti;
        #pragma unroll
        for (int off = 16; off > 0; off >>= 1) {
            float ov = __shfl_xor(v, off, 32);
            int   oi = __shfl_xor(ix, off, 32);
            argmax_combine(v, ix, ov, oi);
        }
        if (lane == 0) { swv[wave] = v; swi[wave] = ix; }
        __syncthreads();
        if (wave == 0) {
            float vv = swv[lane]; int ii = swi[lane];
            #pragma unroll
            for (int off = 16; off > 0; off >>= 1) {
                float ov = __shfl_xor(vv, off, 32);
                int   oi = __shfl_xor(ii, off, 32);
                argmax_combine(vv, ii, ov, oi);
            }
            if (lane == 0) s_win = ii;
        }
        __syncthreads();
        const int pw = s_win;   // next write to s_win is behind a barrier in next iter

        if (tid == 0) r_out[(size_t)b * SPER_ + s] = (float)(base + (size_t)pw);

        const float* xw = x + (base + (size_t)pw) * 3;
        cx = xw[0]; cy = xw[1]; cz = xw[2];

        // ---- fused min-update + next-step local argmax ----
        best = -1.0f; besti = 0;
        #pragma unroll
        for (int i = 0; i < PTS_PER; ++i) {
            int p = tid + i * THREADS_FPS;
            const float* xp = x + (base + (size_t)p) * 3;
            float dx = xp[0] - cx, dy = xp[1] - cy, dz = xp[2] - cz;
            float d = dx * dx + dy * dy + dz * dz;
            float m = mind[i];
            m = fminf(m, d);
            mind[i] = m;
            if (m > best) { best = m; besti = p; }
        }
    }
}

// ---------------------------------------------------------------------------
// Kernel 2: result[o] = sum_s r[s] * W[o][s] + b[o] via V_WMMA_F32_16X16X4_F32.
// A[m][k] = W[obase+m][s0+k], B[k][n] = r[s0+k] broadcast over n.
// 16 blocks (one per 16-output group) x 8 waves, 4 accumulators per wave.
// ---------------------------------------------------------------------------
__global__ __launch_bounds__(256) void matvec_wmma(const float* __restrict__ r,
                                                   const float* __restrict__ W,
                                                   const float* __restrict__ bias,
                                                   float* __restrict__ out) {
    const int obase = blockIdx.x * 16;
    const int tid   = threadIdx.x;
    const int lane  = tid & 31;
    const int wave  = tid >> 5;                 // 0..7
    const int m     = lane & 15;
    const int klo   = (lane < 16) ? 0 : 2;      // A 16x4 f32: VGPR0={K0,K2}, VGPR1={K1,K3}

    __shared__ float partial[8][16];

    const int slice   = S_ / 8;                 // 4096 s-values per wave
    const int s_begin = wave * slice;
    const float* wrow = W + (size_t)(obase + m) * S_;

    v8f acc0 = {}; v8f acc1 = {}; v8f acc2 = {}; v8f acc3 = {};
    for (int s0 = s_begin; s0 < s_begin + slice; s0 += 16) {
        v2f a0 = *(const v2f*)(wrow + s0 + 0  + klo);
        v2f b0 = *(const v2f*)(r    + s0 + 0  + klo);
        acc0 = __builtin_amdgcn_wmma_f32_16x16x4_f32(false, a0, false, b0, (short)0, acc0, false, false);
        v2f a1 = *(const v2f*)(wrow + s0 + 4  + klo);
        v2f b1 = *(const v2f*)(r    + s0 + 4  + klo);
        acc1 = __builtin_amdgcn_wmma_f32_16x16x4_f32(false, a1, false, b1, (short)0, acc1, false, false);
        v2f a2 = *(const v2f*)(wrow + s0 + 8  + klo);
        v2f b2 = *(const v2f*)(r    + s0 + 8  + klo);
        acc2 = __builtin_amdgcn_wmma_f32_16x16x4_f32(false, a2, false, b2, (short)0, acc2, false, false);
        v2f a3 = *(const v2f*)(wrow + s0 + 12 + klo);
        v2f b3 = *(const v2f*)(r    + s0 + 12 + klo);
        acc3 = __builtin_amdgcn_wmma_f32_16x16x4_f32(false, a3, false, b3, (short)0, acc3, false, false);
    }

    v8f acc;
    #pragma unroll
    for (int j = 0; j < 8; ++j) acc[j] = acc0[j] + acc1[j] + acc2[j] + acc3[j];

    // D layout: VGPR j, lanes 0-15 -> M=j (N=lane); lanes 16-31 -> M=j+8.
    // Columns are identical (B broadcast), take N==0 columns: lane 0 and lane 16.
    if (lane == 0) {
        #pragma unroll
        for (int j = 0; j < 8; ++j) partial[wave][j] = acc[j];
    }
    if (lane == 16) {
        #pragma unroll
        for (int j = 0; j < 8; ++j) partial[wave][8 + j] = acc[j];
    }
    __syncthreads();

    if (tid < 16) {
        float sum = bias[obase + tid];
        #pragma unroll
        for (int wv = 0; wv < 8; ++wv) sum += partial[wv][tid];
        out[obase + tid] = sum;
    }
}

// ---------------------------------------------------------------------------
// Kernel 3: loss = mean((result - 1)^2) -> d_out[256]
// ---------------------------------------------------------------------------
__global__ __launch_bounds__(256) void loss_kernel(float* __restrict__ out) {
    __shared__ float red[256];
    const int t = threadIdx.x;
    float v = out[t] - 1.0f;
    red[t] = v * v;
    __syncthreads();
    for (int off = 128; off > 0; off >>= 1) {
        if (t < off) red[t] += red[t + off];
        __syncthreads();
    }
    if (t == 0) out[OUT_] = red[0] / (float)OUT_;
}

extern "C" void kernel_launch(void* const* d_in, const int* in_sizes, int n_in,
                              void* d_out, int out_size, void* d_ws, size_t ws_size,
                              hipStream_t stream) {
    // inputs: 0=x(f32), 1=ptr(i32), 2=ratio(f32), 3=random_start(i32), 4=W(f32), 5=b(f32)
    const float* x    = (const float*)d_in[0];
    const float* W    = (const float*)d_in[4];
    const float* bias = (const float*)d_in[5];
    float*       out  = (float*)d_out;            // 256 result + 1 loss
    float*       r    = (float*)d_ws;             // S_ floats of sampled indices

    fps_kernel  <<<B_,       THREADS_FPS, 0, stream>>>(x, r);
    matvec_wmma <<<OUT_ / 16, 256,        0, stream>>>(r, W, bias, out);
    loss_kernel <<<1,         256,        0, stream>>>(out);
}